// HGNN_16114717294950
// MI455X (gfx1250) — compile-verified
//
#include <hip/hip_runtime.h>

typedef __attribute__((ext_vector_type(2))) float v2f;
typedef __attribute__((ext_vector_type(8))) float v8f;

#define DIMD 128

// ---------------- meta / binning kernels ----------------

__global__ void HGNN_zero_meta(int* meta) {
    if (threadIdx.x < 8) meta[threadIdx.x] = 0;   // counts1[4], counts2[4]
}

__global__ void HGNN_count(const int* __restrict__ etype, int E, int* __restrict__ counts) {
    int i = blockIdx.x * blockDim.x + threadIdx.x;
    if (i < E) atomicAdd(&counts[etype[i]], 1);
}

// meta layout (ints): [0..3] counts1, [4..7] counts2, [8..11] cursor1,
//                     [12..15] cursor2, [16..19] offs1, [20..23] offs2
__global__ void HGNN_scan(int* meta) {
    if (threadIdx.x == 0 && blockIdx.x == 0) {
        int off = 0;
        for (int t = 0; t < 4; ++t) { meta[16 + t] = off; meta[8 + t]  = off; off += meta[t]; }
        off = 0;
        for (int t = 0; t < 4; ++t) { meta[20 + t] = off; meta[12 + t] = off; off += meta[4 + t]; }
    }
}

__global__ void HGNN_fill(const int* __restrict__ etype, int E,
                          int* __restrict__ cursor, int* __restrict__ binned) {
    int i = blockIdx.x * blockDim.x + threadIdx.x;
    if (i < E) {
        int t = etype[i];
        int pos = atomicAdd(&cursor[t], 1);
        binned[pos] = i;
    }
}

__global__ void HGNN_init_out(const float4* __restrict__ x, float4* __restrict__ out, int n4) {
    int i = blockIdx.x * blockDim.x + threadIdx.x;
    if (i < n4) out[i] = x[i];
}

// ---------------- WMMA edge GEMM ----------------
// One wave (32 lanes) handles an EDGES-edge tile of uniform edge-type.
// A: EDGES x K gathered features staged in LDS (padded stride),
// B: W[t] (K x 128) streamed from L2, C/D: EDGES x 128 in v8f register tiles.

template <int ARITY, int EDGES>
__global__ __launch_bounds__(32)
void HGNN_edge_gemm(const float* __restrict__ x,
                    const float* __restrict__ W,      // (4, ARITY*128, 128) row-major
                    const int*   __restrict__ src,    // E * ARITY
                    const int*   __restrict__ tgt,    // E
                    const int*   __restrict__ binned, // E edge ids, grouped by type
                    const int*   __restrict__ counts, // 4
                    const int*   __restrict__ offs,   // 4
                    float*       __restrict__ out,
                    float*       __restrict__ dump) { // 128-float sink for pad rows
    constexpr int K      = ARITY * DIMD;
    constexpr int STRIDE = K + 4;                 // LDS pad: conflict-free A reads
    constexpr int CH     = EDGES / 32;            // edge slots per lane
    constexpr int HT     = EDGES / 16;            // 16-row A tiles
    __shared__ float lds[EDGES * STRIDE];

    const int lane = threadIdx.x;

    // map blockIdx.x -> (edge type t, tile within that type's bin)
    int tile = blockIdx.x;
    int t, cnt = 0;
    for (t = 0; t < 4; ++t) {
        cnt = counts[t];
        int ntiles = (cnt + EDGES - 1) / EDGES;
        if (tile < ntiles) break;
        tile -= ntiles;
    }
    if (t >= 4) return;                           // uniform: surplus blocks exit

    const int base = offs[t] + tile * EDGES;
    const int end  = offs[t] + cnt;

    // per-lane edge metadata (CH slots per lane)
    int e[CH], sA[CH], sB[CH], tg[CH];
#pragma unroll
    for (int c = 0; c < CH; ++c) {
        int slot = base + c * 32 + lane;
        e[c]  = (slot < end) ? binned[slot] : -1;
        tg[c] = (e[c] >= 0) ? tgt[e[c]] : -1;
        sA[c] = (e[c] >= 0) ? src[e[c] * ARITY] : 0;
        sB[c] = 0;
        if (ARITY == 2) sB[c] = (e[c] >= 0) ? src[e[c] * 2 + 1] : 0;
    }

    // stage gathered rows into LDS; coalesced 512B row loads (float4/lane)
    for (int r = 0; r < EDGES; ++r) {
        const int c  = r >> 5;
        const int rl = r & 31;
        const int er = __shfl(e[c], rl);          // uniform per iteration
        float* dst = &lds[r * STRIDE + lane * 4];
        if (er >= 0) {
            const int a = __shfl(sA[c], rl);
            const float4 va = *(const float4*)&x[a * DIMD + lane * 4];
            dst[0] = va.x; dst[1] = va.y; dst[2] = va.z; dst[3] = va.w;
            if (ARITY == 2) {
                const int b = __shfl(sB[c], rl);
                const float4 vb = *(const float4*)&x[b * DIMD + lane * 4];
                float* d2 = dst + DIMD;
                d2[0] = vb.x; d2[1] = vb.y; d2[2] = vb.z; d2[3] = vb.w;
            }
        } else {                                  // pad rows -> zero messages
            dst[0] = 0.f; dst[1] = 0.f; dst[2] = 0.f; dst[3] = 0.f;
            if (ARITY == 2) {
                float* d2 = dst + DIMD;
                d2[0] = 0.f; d2[1] = 0.f; d2[2] = 0.f; d2[3] = 0.f;
            }
        }
    }
    __syncthreads();

    // WMMA fragment coordinates (ISA 16x4 f32 A layout / 4x16 B layout)
    const int m    = lane & 15;                   // row within 16
    const int koff = (lane >> 4) << 1;            // 0 (lanes 0-15) or 2 (lanes 16-31)
    const float* Wt = W + t * K * DIMD;

    v8f acc[HT][8] = {};                          // HT 16-row tiles x 8 N-tiles

    for (int kb = 0; kb < K; kb += 4) {
        v2f a[HT];
#pragma unroll
        for (int h = 0; h < HT; ++h)
            a[h] = *(const v2f*)&lds[(16 * h + m) * STRIDE + kb + koff];
        const float* wrow = &Wt[(kb + koff) * DIMD + m];   // B: K=(kb+koff+v), N=m
#pragma unroll
        for (int nb = 0; nb < 8; ++nb) {
            v2f b;
            b.x = wrow[nb * 16];                  // K = kb+koff
            b.y = wrow[DIMD + nb * 16];           // K = kb+koff+1
#pragma unroll
            for (int h = 0; h < HT; ++h)
                acc[h][nb] = __builtin_amdgcn_wmma_f32_16x16x4_f32(
                    false, a[h], false, b, (short)0, acc[h][nb], false, false);
        }
    }

    // branchless scatter-add: pad rows are zero and go to the dump sink,
    // so every atomic is unconditional (no exec-mask churn).
    const int nlane = lane & 15;
    const int mhalf = (lane >> 4) << 3;           // 0 or 8
#pragma unroll
    for (int h = 0; h < HT; ++h) {
#pragma unroll
        for (int r = 0; r < 8; ++r) {
            // edge row = h*16 + r + mhalf; r+mhalf <= 15 so chunk = h/2
            const int tgr = __shfl(tg[h >> 1], ((h & 1) << 4) + r + mhalf);
            float* po = (tgr >= 0) ? (out + (size_t)tgr * DIMD) : dump;
#pragma unroll
            for (int nb = 0; nb < 8; ++nb)
                atomicAdd(&po[nb * 16 + nlane], acc[h][nb][r]);
        }
    }
}

// ---------------- launch ----------------

extern "C" void kernel_launch(void* const* d_in, const int* in_sizes, int n_in,
                              void* d_out, int out_size, void* d_ws, size_t ws_size,
                              hipStream_t stream) {
    const float* x     = (const float*)d_in[0];
    const float* W1    = (const float*)d_in[1];
    const float* W2    = (const float*)d_in[2];
    const int*   src1  = (const int*)d_in[3];
    const int*   tgt1  = (const int*)d_in[4];
    const int*   type1 = (const int*)d_in[5];
    const int*   src2  = (const int*)d_in[6];
    const int*   tgt2  = (const int*)d_in[7];
    const int*   type2 = (const int*)d_in[8];
    float*       out   = (float*)d_out;

    const int E1     = in_sizes[3];          // 500000
    const int E2     = in_sizes[7];          // 250000
    const int nElems = in_sizes[0];          // N_NODES * 128

    int* meta    = (int*)d_ws;
    int* counts1 = meta;
    int* counts2 = meta + 4;
    int* cursor1 = meta + 8;
    int* cursor2 = meta + 12;
    int* offs1   = meta + 16;
    int* offs2   = meta + 20;
    int* binned1 = meta + 32;
    int* binned2 = binned1 + E1;
    float* dump  = (float*)(binned2 + E2);   // 128-float atomic sink

    HGNN_zero_meta<<<1, 32, 0, stream>>>(meta);
    HGNN_count<<<(E1 + 255) / 256, 256, 0, stream>>>(type1, E1, counts1);
    HGNN_count<<<(E2 + 255) / 256, 256, 0, stream>>>(type2, E2, counts2);
    HGNN_scan<<<1, 1, 0, stream>>>(meta);
    HGNN_fill<<<(E1 + 255) / 256, 256, 0, stream>>>(type1, E1, cursor1, binned1);
    HGNN_fill<<<(E2 + 255) / 256, 256, 0, stream>>>(type2, E2, cursor2, binned2);

    HGNN_init_out<<<(nElems / 4 + 255) / 256, 256, 0, stream>>>(
        (const float4*)x, (float4*)out, nElems / 4);

    HGNN_edge_gemm<1, 64><<<(E1 + 63) / 64 + 4, 32, 0, stream>>>(
        x, W1, src1, tgt1, binned1, counts1, offs1, out, dump);
    HGNN_edge_gemm<2, 32><<<(E2 + 31) / 32 + 4, 32, 0, stream>>>(
        x, W2, src2, tgt2, binned2, counts2, offs2, out, dump);
}